// HeteroGCNEncoder_26774826123587
// MI455X (gfx1250) — compile-verified
//
#include <hip/hip_runtime.h>

typedef __attribute__((ext_vector_type(2))) float v2f;
typedef __attribute__((ext_vector_type(8))) float v8f;

#define N_TICKER 5000
#define N_INST   20000
#define N_MF     30000
#define N_NEWS   100000
#define NE       1000000

__global__ void zero_f32(float* __restrict__ p, int n) {
  int i = blockIdx.x * blockDim.x + threadIdx.x;
  if (i < n) p[i] = 0.0f;
}

// degree count per destination node (float so later scale is a single div)
__global__ void count_edges(const int* __restrict__ ei, float* __restrict__ cnt, int nE) {
  int e = blockIdx.x * blockDim.x + threadIdx.x;
  if (e < nE) atomicAdd(&cnt[ei[nE + e]], 1.0f);
}

__global__ void sum3_f32(const float* __restrict__ a, const float* __restrict__ b,
                         const float* __restrict__ c, float* __restrict__ o, int n) {
  int i = blockIdx.x * blockDim.x + threadIdx.x;
  if (i < n) o[i] = a[i] + b[i] + c[i];
}

// C[M,64] = A[M,K] @ B[K,64] (+ bias[64]); all row-major f32.
// One wave32 produces a 16x64 strip via 4 accumulators of V_WMMA_F32_16X16X4_F32.
__global__ void gemm_n64(const float* __restrict__ A, const float* __restrict__ B,
                         const float* __restrict__ bias, float* __restrict__ C,
                         int M, int K) {
  const int lane = threadIdx.x & 31;
  const int wave = (int)((blockIdx.x * blockDim.x + threadIdx.x) >> 5);
  const int row0 = wave * 16;
  if (row0 >= M) return;
  const int half = lane >> 4;   // 0: lanes 0-15, 1: lanes 16-31
  const int l16  = lane & 15;

  int arow = row0 + l16;
  if (arow >= M) arow = M - 1;  // clamped read; per-row guarded store below
  const float* Arow = A + (size_t)arow * K;

  v8f acc0 = {}, acc1 = {}, acc2 = {}, acc3 = {};
  for (int k0 = 0; k0 < K; k0 += 4) {
    const int ka = k0 + 2 * half;              // this half-wave holds K pair (ka, ka+1)
    v2f a = *(const v2f*)(Arow + ka);          // A[row][ka], A[row][ka+1]
    const float* B0 = B + (size_t)ka * 64;     // row ka of B
    const float* B1 = B0 + 64;                 // row ka+1 of B
    v2f b0, b1, b2, b3;
    b0.x = B0[l16 +  0]; b0.y = B1[l16 +  0];
    b1.x = B0[l16 + 16]; b1.y = B1[l16 + 16];
    b2.x = B0[l16 + 32]; b2.y = B1[l16 + 32];
    b3.x = B0[l16 + 48]; b3.y = B1[l16 + 48];
    acc0 = __builtin_amdgcn_wmma_f32_16x16x4_f32(false, a, false, b0, (short)0, acc0, false, false);
    acc1 = __builtin_amdgcn_wmma_f32_16x16x4_f32(false, a, false, b1, (short)0, acc1, false, false);
    acc2 = __builtin_amdgcn_wmma_f32_16x16x4_f32(false, a, false, b2, (short)0, acc2, false, false);
    acc3 = __builtin_amdgcn_wmma_f32_16x16x4_f32(false, a, false, b3, (short)0, acc3, false, false);
  }

  float bv0 = bias ? bias[l16 +  0] : 0.0f;
  float bv1 = bias ? bias[l16 + 16] : 0.0f;
  float bv2 = bias ? bias[l16 + 32] : 0.0f;
  float bv3 = bias ? bias[l16 + 48] : 0.0f;

  #pragma unroll
  for (int r = 0; r < 8; ++r) {
    int rowg = row0 + r + 8 * half;            // C/D layout: VGPR r -> row r (lanes 0-15) / r+8
    if (rowg < M) {
      float* Crow = C + (size_t)rowg * 64;
      Crow[l16 +  0] = acc0[r] + bv0;
      Crow[l16 + 16] = acc1[r] + bv1;
      Crow[l16 + 32] = acc2[r] + bv2;
      Crow[l16 + 48] = acc3[r] + bv3;
    }
  }
}

// One wave per edge: out[dst][0:64] += y[src][0:64] / max(cnt[dst],1)
__global__ void scatter_mean(const float* __restrict__ y, const int* __restrict__ ei,
                             const float* __restrict__ cnt, float* __restrict__ out, int nE) {
  int gid  = blockIdx.x * blockDim.x + threadIdx.x;
  int e    = gid >> 5;
  int lane = gid & 31;
  if (e >= nE) return;
  int src = ei[e];
  int dst = ei[nE + e];
  float scale = 1.0f / fmaxf(cnt[dst], 1.0f);
  const float* ys = y   + (size_t)src * 64;
  float*       od = out + (size_t)dst * 64;
  atomicAdd(&od[lane],      ys[lane]      * scale);
  atomicAdd(&od[lane + 32], ys[lane + 32] * scale);
}

extern "C" void kernel_launch(void* const* d_in, const int* in_sizes, int n_in,
                              void* d_out, int out_size, void* d_ws, size_t ws_size,
                              hipStream_t stream) {
  const float* x_t = (const float*)d_in[0];
  const float* x_i = (const float*)d_in[1];
  const float* x_m = (const float*)d_in[2];
  const float* x_n = (const float*)d_in[3];

  const int   *ei_hit = (const int*)d_in[4],  *ei_hmt = (const int*)d_in[8],
              *ei_ant = (const int*)d_in[12], *ei_rmt = (const int*)d_in[16],
              *ei_rit = (const int*)d_in[20], *ei_ran = (const int*)d_in[24];
  const float *Wl_hit = (const float*)d_in[5],  *bl_hit = (const float*)d_in[6],  *Wr_hit = (const float*)d_in[7];
  const float *Wl_hmt = (const float*)d_in[9],  *bl_hmt = (const float*)d_in[10], *Wr_hmt = (const float*)d_in[11];
  const float *Wl_ant = (const float*)d_in[13], *bl_ant = (const float*)d_in[14], *Wr_ant = (const float*)d_in[15];
  const float *Wl_rmt = (const float*)d_in[17], *bl_rmt = (const float*)d_in[18], *Wr_rmt = (const float*)d_in[19];
  const float *Wl_rit = (const float*)d_in[21], *bl_rit = (const float*)d_in[22], *Wr_rit = (const float*)d_in[23];
  const float *Wl_ran = (const float*)d_in[25], *bl_ran = (const float*)d_in[26], *Wr_ran = (const float*)d_in[27];

  // ---- workspace layout (floats) ----
  float* ws      = (float*)d_ws;
  float* y_hit   = ws;                          // ticker      @ Wl_hit : 5000*64
  float* y_hmt   = y_hit + N_TICKER * 64;       // ticker      @ Wl_hmt : 5000*64
  float* y_ant   = y_hmt + N_TICKER * 64;       // news        @ Wl_ant : 100000*64
  float* y_rmt   = y_ant + N_NEWS   * 64;       // mutual_fund @ Wl_rmt : 30000*64
  float* y_rit   = y_rmt + N_MF     * 64;       // institution @ Wl_rit : 20000*64
  float* y_ran   = y_rit + N_INST   * 64;       // ticker      @ Wl_ran : 5000*64
  float* cnt_hit = y_ran + N_TICKER * 64;       // 20000
  float* cnt_hmt = cnt_hit + N_INST;            // 30000
  float* cnt_ant = cnt_hmt + N_MF;              // 5000
  float* cnt_rmt = cnt_ant + N_TICKER;          // 5000
  float* cnt_rit = cnt_rmt + N_TICKER;          // 5000
  float* cnt_ran = cnt_rit + N_TICKER;          // 100000
  float* WrSum   = cnt_ran + N_NEWS;            // 64*64
  float* blSum   = WrSum + 64 * 64;             // 64

  const int cntTotal = N_INST + N_MF + 3 * N_TICKER + N_NEWS;  // 165000

  // ---- 1) zero degree counters, count degrees ----
  zero_f32<<<(cntTotal + 255) / 256, 256, 0, stream>>>(cnt_hit, cntTotal);
  count_edges<<<(NE + 255) / 256, 256, 0, stream>>>(ei_hit, cnt_hit, NE);
  count_edges<<<(NE + 255) / 256, 256, 0, stream>>>(ei_hmt, cnt_hmt, NE);
  count_edges<<<(NE + 255) / 256, 256, 0, stream>>>(ei_ant, cnt_ant, NE);
  count_edges<<<(NE + 255) / 256, 256, 0, stream>>>(ei_rmt, cnt_rmt, NE);
  count_edges<<<(NE + 255) / 256, 256, 0, stream>>>(ei_rit, cnt_rit, NE);
  count_edges<<<(NE + 255) / 256, 256, 0, stream>>>(ei_ran, cnt_ran, NE);

  // ---- 2) transform sources: y = x_src @ Wl (WMMA f32) ----
  auto gemm = [&](const float* A, const float* B, const float* bias, float* C, int M, int K) {
    int waves  = (M + 15) / 16;
    int blocks = (waves + 7) / 8;               // 256 threads = 8 wave32 per block
    gemm_n64<<<blocks, 256, 0, stream>>>(A, B, bias, C, M, K);
  };
  gemm(x_t, Wl_hit, nullptr, y_hit, N_TICKER, 64);
  gemm(x_t, Wl_hmt, nullptr, y_hmt, N_TICKER, 64);
  gemm(x_n, Wl_ant, nullptr, y_ant, N_NEWS,   128);
  gemm(x_m, Wl_rmt, nullptr, y_rmt, N_MF,     64);
  gemm(x_i, Wl_rit, nullptr, y_rit, N_INST,   64);
  gemm(x_t, Wl_ran, nullptr, y_ran, N_TICKER, 64);

  // ---- 3) ticker gets three lin_r terms with the same input: pre-sum Wr / bl ----
  sum3_f32<<<(64 * 64 + 255) / 256, 256, 0, stream>>>(Wr_ant, Wr_rmt, Wr_rit, WrSum, 64 * 64);
  sum3_f32<<<1, 64, 0, stream>>>(bl_ant, bl_rmt, bl_rit, blSum, 64);

  // ---- 4) init outputs: out = x_dst @ Wr + bl  (then scatter accumulates on top) ----
  float* out   = (float*)d_out;
  float* out_t = out;                                   // [5000,64]
  float* out_i = out_t + N_TICKER * 64;                 // [20000,64]
  float* out_m = out_i + N_INST * 64;                   // [30000,64]
  float* out_n = out_m + N_MF * 64;                     // [100000,64]
  gemm(x_t, WrSum,  blSum,  out_t, N_TICKER, 64);
  gemm(x_i, Wr_hit, bl_hit, out_i, N_INST,   64);
  gemm(x_m, Wr_hmt, bl_hmt, out_m, N_MF,     64);
  gemm(x_n, Wr_ran, bl_ran, out_n, N_NEWS,   128);

  // ---- 5) scaled atomic scatter: out[dst] += y[src] / max(cnt[dst],1) ----
  auto scat = [&](const float* y, const int* ei, const float* cnt, float* o) {
    long threads = (long)NE * 32;
    scatter_mean<<<(int)((threads + 255) / 256), 256, 0, stream>>>(y, ei, cnt, o, NE);
  };
  scat(y_hit, ei_hit, cnt_hit, out_i);
  scat(y_hmt, ei_hmt, cnt_hmt, out_m);
  scat(y_ant, ei_ant, cnt_ant, out_t);
  scat(y_rmt, ei_rmt, cnt_rmt, out_t);
  scat(y_rit, ei_rit, cnt_rit, out_t);
  scat(y_ran, ei_ran, cnt_ran, out_n);
}